// GIN_2877628089017
// MI455X (gfx1250) — compile-verified
//
#include <hip/hip_runtime.h>

// ---------------------------------------------------------------------------
// GIN 2-layer forward for gfx1250 (MI455X).
//   agg  = scatter_add(feat[src] -> dst);  h = relu((feat+agg) @ W1 + b1)
//   agg2 = scatter_add(h[src]   -> dst);   out = (h+agg2) @ W2 + b2
// GEMMs use V_WMMA_F32_16X16X4_F32 (full fp32 precision; GEMM compute ~5 GFLOP
// is negligible vs the L2-resident edge-scatter phase, so no quantization).
// W is staged in LDS with K-row pairs interleaved per column so every B
// fragment is a single aligned ds_load_b64 into an even VGPR pair.
// ---------------------------------------------------------------------------

typedef __attribute__((ext_vector_type(2))) float v2f;
typedef __attribute__((ext_vector_type(8))) float v8f;

#define N_FEAT 128

// buf = src  (vectorized float4 copy)
__global__ void copy_f4(const float4* __restrict__ in, float4* __restrict__ out, int n) {
  int i = blockIdx.x * blockDim.x + threadIdx.x;
  if (i < n) out[i] = in[i];
}

// accum[dst[e]] += feat[src[e]]  — one wave32 per edge, float4 slice per lane,
// 4 x global_atomic_add_f32 per lane (lands in L2 atomic units; node array is
// 51.2MB and L2 is 192MB, so this phase is L2-resident).
__global__ void scatter_add_edges(const float* __restrict__ feat,
                                  const int* __restrict__ src,
                                  const int* __restrict__ dst,
                                  float* accum, int n_edges) {
  int gid = blockIdx.x * blockDim.x + threadIdx.x;
  int e = gid >> 5;                 // wave-aligned: all 32 lanes share one edge
  if (e >= n_edges) return;
  int lane = gid & 31;
  int s = src[e];
  int d = dst[e];
  const float4 v = ((const float4*)(feat + (size_t)s * N_FEAT))[lane];
  float* p = accum + (size_t)d * N_FEAT + (lane << 2);
  atomicAdd(p + 0, v.x);
  atomicAdd(p + 1, v.y);
  atomicAdd(p + 2, v.z);
  atomicAdd(p + 3, v.w);
}

// out0[m,n] (= out1[m,n]) = act( sum_k A[m,k]*W[k,n] + bias[n] )
// Block: 256 thr = 8 waves; each wave computes a 16 x NOUT slab via
// V_WMMA_F32_16X16X4_F32, all NOUT/16 accumulators live so A is read once.
// Requires M % 16 == 0 (true: 100000).
template <int NOUT, bool RELU, bool DUAL>
__global__ __launch_bounds__(256)
void gin_gemm(const float* A, const float* __restrict__ W,
              const float* __restrict__ bias, float* out0, float* out1, int M) {
  constexpr int NT = NOUT / 16;
  // Pair-interleaved W: Wl[(k>>1)*2*NOUT + col*2 + (k&1)]  => the B fragment
  // {W[k][col], W[k+1][col]} is 8B-contiguous & aligned -> one ds_load_b64.
  __shared__ float Wl[N_FEAT * NOUT];
  const int tid = threadIdx.x;

  for (int p = tid; p < (N_FEAT / 2) * NOUT; p += 256) {
    const int kp  = p / NOUT;        // K-row pair index
    const int col = p % NOUT;
    v2f w;
    w.x = W[(2 * kp + 0) * NOUT + col];
    w.y = W[(2 * kp + 1) * NOUT + col];
    *(v2f*)&Wl[kp * 2 * NOUT + col * 2] = w;   // ds_store_b64
  }
  __syncthreads();

  const int wave = tid >> 5;
  const int lane = tid & 31;
  const int rowbase = blockIdx.x * 128 + wave * 16;
  if (rowbase >= M) return;          // whole-wave granularity (M % 16 == 0)

  const int lrow  = lane & 15;       // M (for A) / N (for B) within tile
  const int khalf = (lane >> 4) << 1;// lanes 0-15: K=k..k+1, lanes 16-31: K=k+2..k+3

  const float* Arow = A + (size_t)(rowbase + lrow) * N_FEAT + khalf;
  // per-lane base into the pair-interleaved W: pair row (khalf>>1), column lrow
  const float* Wbase = &Wl[(khalf >> 1) * 2 * NOUT + lrow * 2];

  v8f acc[NT] = {};

#pragma unroll 4
  for (int k = 0; k < N_FEAT; k += 4) {
    // A fragment: 16x4 fp32 tile, 2 VGPRs/lane (ISA 7.12.2 layout)
    v2f a = *(const v2f*)(Arow + k);
    const float* wk = Wbase + (k >> 1) * 2 * NOUT;  // advance (k/2) pair rows
#pragma unroll
    for (int t = 0; t < NT; ++t) {
      // B fragment: {W[k+khalf][t*16+lrow], W[k+khalf+1][t*16+lrow]} -> b64
      v2f b = *(const v2f*)(wk + t * 32);
      acc[t] = __builtin_amdgcn_wmma_f32_16x16x4_f32(
          /*neg_a=*/false, a, /*neg_b=*/false, b,
          /*c_mod=*/(short)0, acc[t], /*reuse_a=*/false, /*reuse_b=*/false);
    }
  }

  // C/D layout: VGPR r -> M = r (+8 for lanes 16-31), N = lane%16
  const int srowoff = (lane >> 4) << 3;
#pragma unroll
  for (int t = 0; t < NT; ++t) {
    const int col = t * 16 + lrow;
    const float bval = bias[col];
#pragma unroll
    for (int r = 0; r < 8; ++r) {
      const int row = rowbase + srowoff + r;
      float v = acc[t][r] + bval;
      if (RELU) v = fmaxf(v, 0.0f);
      out0[(size_t)row * NOUT + col] = v;
      if (DUAL) out1[(size_t)row * NOUT + col] = v;
    }
  }
}

extern "C" void kernel_launch(void* const* d_in, const int* in_sizes, int n_in,
                              void* d_out, int out_size, void* d_ws, size_t ws_size,
                              hipStream_t stream) {
  const float* feat = (const float*)d_in[0];
  const int*   src  = (const int*)d_in[1];
  const int*   dst  = (const int*)d_in[2];
  const float* W1   = (const float*)d_in[3];
  const float* b1   = (const float*)d_in[4];
  const float* W2   = (const float*)d_in[5];
  const float* b2   = (const float*)d_in[6];
  float* out = (float*)d_out;

  const int N = in_sizes[0] / N_FEAT;   // 100000
  const int E = in_sizes[1];            // 1600000

  float* buf0 = (float*)d_ws;                       // accumulation target
  float* buf1 = buf0 + (size_t)N * N_FEAT;          // gather source (x)

  // 1) buf0 = features
  {
    int n4 = (N * N_FEAT) / 4;
    copy_f4<<<(n4 + 255) / 256, 256, 0, stream>>>(
        (const float4*)feat, (float4*)buf0, n4);
  }
  // 2) buf0[dst] += feat[src]
  {
    long long t = (long long)E * 32;
    scatter_add_edges<<<(int)((t + 255) / 256), 256, 0, stream>>>(
        feat, src, dst, buf0, E);
  }
  // 3) buf1 = relu(buf0 @ W1 + b1); buf0 = buf1 (init for next scatter)
  gin_gemm<128, true, true><<<(N + 127) / 128, 256, 0, stream>>>(
      buf0, W1, b1, buf1, buf0, N);
  // 4) buf0[dst] += buf1[src]
  {
    long long t = (long long)E * 32;
    scatter_add_edges<<<(int)((t + 255) / 256), 256, 0, stream>>>(
        buf1, src, dst, buf0, E);
  }
  // 5) out = buf0 @ W2 + b2
  gin_gemm<64, false, false><<<(N + 127) / 128, 256, 0, stream>>>(
      buf0, W2, b2, out, nullptr, N);
}